// Net_CL_54511724920980
// MI455X (gfx1250) — compile-verified
//
#include <hip/hip_runtime.h>
#include <math.h>

// ---- problem constants (from the reference) ----
constexpr int Bn   = 64;     // batches
constexpr int Dn   = 256;    // rows
constexpr int Mn   = 100;    // features
constexpr int FTS  = 257;    // transposed LDS row stride (odd dwords -> conflict-free)
constexpr int NTRI = Dn * (Dn + 1) / 2;   // 32896 triu entries per batch
constexpr int CHUNKS    = 8;              // blocks per batch in phase 2
constexpr int PER_CHUNK = NTRI / CHUNKS;  // 4112 (exact)

typedef __attribute__((ext_vector_type(2))) float v2f;
typedef __attribute__((ext_vector_type(8))) float v8f;

__device__ __forceinline__ int triBase(int i) {
    // start of row i in flat triu order: i*D - i*(i-1)/2
    return i * Dn - ((i * (i - 1)) >> 1);
}

// ---------------------------------------------------------------------------
// Phase 1: row sums r[b,i] = 0.5 * sum_j sum_m (|fi+fj| - |fi-fj|)
// One block = half a batch (8 row-strips of 16). Each wave owns one strip and
// accumulates S(strip,:) x Ones via V_WMMA_F32_16X16X4_F32, computing the S
// elements directly in the WMMA A-matrix register layout.
// ---------------------------------------------------------------------------
__global__ __launch_bounds__(256) void rowsum_kernel(const float* __restrict__ feat,
                                                     float* __restrict__ rs) {
    extern __shared__ float sm[];
    float* Ft = sm;                               // Ft[m*FTS + d], transposed feat
    const int bid     = blockIdx.x;
    const int batch   = bid >> 1;
    const int halfsel = bid & 1;
    const int tid     = threadIdx.x;
    const int wave    = tid >> 5;
    const int lane    = tid & 31;

    // Stage feat[batch] transposed into LDS. Global reads coalesced; LDS write
    // stride 257 dwords -> consecutive banks, conflict-free.
    const float* fb = feat + (size_t)batch * (Dn * Mn);
    for (int idx = tid; idx < Dn * Mn; idx += 256) {   // exactly 100 iters, no divergence
        int d = idx / Mn;
        int m = idx - d * Mn;
        Ft[m * FTS + d] = fb[idx];
    }
    __syncthreads();

    const int strip = halfsel * 8 + wave;   // 0..15
    const int i0    = strip * 16;
    const int Mr    = lane & 15;            // A-matrix M = lane % 16
    const int kh    = (lane >> 4) << 1;     // A-matrix K base: 0 (lanes<16) or 2
    const int irow  = i0 + Mr;

    v8f c = {0.f, 0.f, 0.f, 0.f, 0.f, 0.f, 0.f, 0.f};
    v2f ones;
    ones.x = 1.0f; ones.y = 1.0f;           // B = 4x16 all-ones (layout-immune)

    for (int k0 = 0; k0 < Dn; k0 += 4) {    // 64 WMMA accumulation steps
        const int j0 = k0 + kh;
        float a0 = 0.f, a1 = 0.f;
        #pragma unroll 4
        for (int m = 0; m < Mn; ++m) {
            float fi  = Ft[m * FTS + irow];     // broadcast-friendly (stride 1 in lane)
            float fj0 = Ft[m * FTS + j0];       // 2 distinct addrs across wave
            float fj1 = Ft[m * FTS + j0 + 1];
            a0 += fabsf(fi + fj0) - fabsf(fi - fj0);
            a1 += fabsf(fi + fj1) - fabsf(fi - fj1);
        }
        v2f a;                   // VGPR0 -> K=kh (j0), VGPR1 -> K=kh+1 (j0+1)
        a.x = a0; a.y = a1;
        // D(16x16,f32) += A(16x4,f32) x Ones(4x16,f32); EXEC is all ones here.
        c = __builtin_amdgcn_wmma_f32_16x16x4_f32(false, a, false, ones,
                                                  (short)0, c, false, false);
    }

    // C/D layout: VGPR v, lanes 0-15 -> M=v, lanes 16-31 -> M=v+8; every N column
    // holds the same full row sum. Lanes with N==0 (lane 0 and 16) write 8 rows.
    if ((lane & 15) == 0) {
        const int mb = i0 + ((lane >> 4) << 3);
        #pragma unroll
        for (int v = 0; v < 8; ++v)
            rs[batch * Dn + mb + v] = 0.5f * c[v];
    }
}

// ---------------------------------------------------------------------------
// Phase 2: emit triu entries out[b,p] = e*S(i,j) - (r[i]+r[j])*e/256,
// p in row-major triu order (matches np.triu_indices gather in the reference).
// ---------------------------------------------------------------------------
__global__ __launch_bounds__(256) void triu_kernel(const float* __restrict__ feat,
                                                   const float* __restrict__ temp,
                                                   const float* __restrict__ rs,
                                                   float* __restrict__ out) {
    extern __shared__ float sm[];
    float* Ft = sm;                 // Ft[m*FTS + d]
    float* Rs = sm + Mn * FTS;      // 256 row sums
    const int batch = blockIdx.x / CHUNKS;
    const int chunk = blockIdx.x - batch * CHUNKS;
    const int tid   = threadIdx.x;

    const float* fb = feat + (size_t)batch * (Dn * Mn);
    for (int idx = tid; idx < Dn * Mn; idx += 256) {
        int d = idx / Mn;
        int m = idx - d * Mn;
        Ft[m * FTS + d] = fb[idx];
    }
    Rs[tid] = rs[batch * Dn + tid];
    __syncthreads();

    const float e   = expf(temp[0]);
    const float esc = e * (1.0f / (float)Dn);
    const int pend  = (chunk + 1) * PER_CHUNK;

    for (int p = chunk * PER_CHUNK + tid; p < pend; p += 256) {
        // invert flat triu index: i = largest row with triBase(i) <= p
        // (2D+1)=513, (2D+1)^2 = 263169
        int i = (int)((513.0f - sqrtf((float)(263169 - 8 * p))) * 0.5f);
        if (i < 0) i = 0;
        if (i > Dn - 1) i = Dn - 1;
        while (triBase(i + 1) <= p) ++i;    // exact integer fixup
        while (triBase(i) > p) --i;
        const int j = i + (p - triBase(i));

        float acc = 0.f;
        #pragma unroll 4
        for (int m = 0; m < Mn; ++m) {
            float fi = Ft[m * FTS + i];   // broadcast across wave (i mostly uniform)
            float fj = Ft[m * FTS + j];   // consecutive j -> consecutive banks
            acc += fabsf(fi + fj) - fabsf(fi - fj);
        }
        out[(size_t)batch * NTRI + p] = 0.5f * acc * e - (Rs[i] + Rs[j]) * esc;
    }
}

// ---------------------------------------------------------------------------
extern "C" void kernel_launch(void* const* d_in, const int* in_sizes, int n_in,
                              void* d_out, int out_size, void* d_ws, size_t ws_size,
                              hipStream_t stream) {
    const float* feat = (const float*)d_in[0];   // (64,256,100) f32
    const float* temp = (const float*)d_in[1];   // (1,1) f32
    float*       rsum = (float*)d_ws;            // 64*256 f32 scratch (64 KB)
    float*       out  = (float*)d_out;           // 64*32896 f32

    const size_t shmem1 = (size_t)(Mn * FTS) * sizeof(float);          // 102800 B
    const size_t shmem2 = shmem1 + (size_t)Dn * sizeof(float);         // +1 KB

    rowsum_kernel<<<dim3(Bn * 2),      dim3(256), shmem1, stream>>>(feat, rsum);
    triu_kernel <<<dim3(Bn * CHUNKS), dim3(256), shmem2, stream>>>(feat, temp, rsum, out);
}